// MatchSum_46394236731635
// MI455X (gfx1250) — compile-verified
//
#include <hip/hip_runtime.h>
#include <math.h>

// ---------------- problem constants (from setup_inputs) ----------------
#define B_   4
#define L_   512
#define LS_  128
#define C_   20
#define LC_  128
#define H_   768
#define R_   512
#define LG_  1019          // gram rows per batch: 510 (y1) + 509 (y2)

typedef __attribute__((ext_vector_type(2))) float v2f;
typedef __attribute__((ext_vector_type(8))) float v8f;

#define SQRT_C_F 0.94868329805051381f   // sqrt(0.9)
#define CURV_F   0.9f

// pdist(x, y) = 2/sqrt(c) * atanh(clip(sqrt(c)*|mobius_add(-x,y)|)) expressed
// purely in terms of x2=|x|^2, y2=|y|^2, xy=<x,y>  (mobius_add(-x,y):
//   alpha = 1 - 2c*xy + c*y2 (coeff on -x), beta = 1 - c*x2 (coeff on y))
__device__ __forceinline__ float pdist_from(float x2, float y2, float xy) {
  const float c  = CURV_F;
  const float sc = SQRT_C_F;
  float alpha = 1.0f - 2.0f * c * xy + c * y2;
  float beta  = 1.0f - c * x2;
  float num2  = alpha * alpha * x2 - 2.0f * alpha * beta * xy + beta * beta * y2;
  float den   = fmaxf(1.0f - 2.0f * c * xy + c * c * x2 * y2, 1e-15f);
  float d2    = num2 / (den * den);
  float n     = sqrtf(fmaxf(d2, 1e-30f));
  float z     = fminf(sc * n, 1.0f - 1e-7f);
  return (2.0f / sc) * atanhf(z);
}

// ---------------- max-reduce over length dim ----------------
// in: [ngroups][rows][width] -> out: [ngroups][width] (max over rows)
__global__ void max_reduce_kernel(const float* __restrict__ in, float* __restrict__ out,
                                  int ngroups, int rows, int width) {
  int idx = blockIdx.x * blockDim.x + threadIdx.x;
  int total = ngroups * width;
  if (idx >= total) return;
  int g = idx / width;
  int h = idx - g * width;
  const float* p = in + (size_t)g * rows * width + h;
  float m = p[0];
  for (int l = 1; l < rows; ++l) m = fmaxf(m, p[(size_t)l * width]);
  out[idx] = m;
}

// ---------------- generic fp32 WMMA GEMM ----------------
// C[b] (M x N) = A[b] (M x K, row stride lda) @ B (K x N) [+ bias] [relu]
// One wave computes one 16x16 C tile via V_WMMA_F32_16X16X4_F32.
// 4 waves per block cover 4 consecutive N tiles (64 cols). N must be %64.
// K must be %4. Rows clamp-loaded / guard-stored, so any M works.
__global__ __launch_bounds__(128)
void wmma_gemm_f32(const float* __restrict__ A, long long Abatch, int lda,
                   const float* __restrict__ B, int ldb,
                   const float* __restrict__ bias,
                   float* __restrict__ C, long long Cbatch, int ldc,
                   int M, int K, int relu) {
  int wave = threadIdx.x >> 5;
  int lane = threadIdx.x & 31;
  int l16  = lane & 15;
  int kh   = (lane >> 4) << 1;             // 0 for lanes 0-15, 2 for 16-31
  int m0   = blockIdx.x * 16;
  int n0   = (blockIdx.y * 4 + wave) * 16;
  const float* Ab = A + (size_t)blockIdx.z * (size_t)Abatch;
  float*       Cb = C + (size_t)blockIdx.z * (size_t)Cbatch;

  int rowA = m0 + l16;
  if (rowA > M - 1) rowA = M - 1;          // clamp (rows always valid)
  const float* Ap = Ab + (size_t)rowA * lda + kh;
  const float* Bp = B + (size_t)kh * ldb + n0 + l16;

  v8f acc = {0.f, 0.f, 0.f, 0.f, 0.f, 0.f, 0.f, 0.f};
  int steps = K >> 2;
#pragma unroll 4
  for (int s = 0; s < steps; ++s) {
    v2f a, bv;
    a[0]  = Ap[0];                          // A[row][k0+kh]
    a[1]  = Ap[1];                          // A[row][k0+kh+1]
    bv[0] = Bp[0];                          // B[k0+kh][col]
    bv[1] = Bp[(size_t)ldb];                // B[k0+kh+1][col]
    acc = __builtin_amdgcn_wmma_f32_16x16x4_f32(
        false, a, false, bv, (short)0, acc, false, false);
    Ap += 4;
    Bp += (size_t)4 * ldb;
  }

  int colC  = n0 + l16;
  float bv0 = bias ? bias[colC] : 0.0f;
  int rbase = m0 + ((lane >> 4) << 3);      // +8 for high half-lanes
#pragma unroll
  for (int i = 0; i < 8; ++i) {
    int r = rbase + i;
    if (r < M) {
      float v = acc[i] + bv0;
      if (relu) v = fmaxf(v, 0.0f);
      Cb[(size_t)r * ldc + colC] = v;
    }
  }
}

// ---------------- expmap0 (row scale) ----------------
// out_row = s * in_row; also emits |out_row|^2 per row.
__global__ __launch_bounds__(256)
void expmap0_kernel(const float* __restrict__ in, float* __restrict__ out,
                    float* __restrict__ n2out, int ncols) {
  __shared__ float red[256];
  int row = blockIdx.x;
  int t = threadIdx.x;
  const float* u = in + (size_t)row * ncols;
  float* o = out + (size_t)row * ncols;
  float ss = 0.0f;
  for (int k = t; k < ncols; k += 256) { float v = u[k]; ss = fmaf(v, v, ss); }
  red[t] = ss;
  __syncthreads();
  for (int off = 128; off > 0; off >>= 1) {
    if (t < off) red[t] += red[t + off];
    __syncthreads();
  }
  float u2 = red[0];
  float n  = fmaxf(sqrtf(fmaxf(u2, 1e-30f)), 1e-15f);
  float s  = tanhf(SQRT_C_F * n) / (SQRT_C_F * n);
  float gn = fmaxf(sqrtf(fmaxf(s * s * u2, 1e-30f)), 1e-15f);
  const float maxnorm = (1.0f - 0.004f) / SQRT_C_F;
  if (gn > maxnorm) s *= maxnorm / gn;
  for (int k = t; k < ncols; k += 256) o[k] = u[k] * s;
  if (t == 0 && n2out) n2out[row] = s * s * u2;
}

// ---------------- per-token distances (1/pdist against 22 targets) -------
__global__ __launch_bounds__(256)
void distances_kernel(const float* __restrict__ dwe,  const float* __restrict__ dwe_n2,
                      const float* __restrict__ doce, const float* __restrict__ doc_n2,
                      const float* __restrict__ sume, const float* __restrict__ sum_n2,
                      const float* __restrict__ cande, const float* __restrict__ cand_n2,
                      float* __restrict__ doc_inter, float* __restrict__ sum_inter,
                      float* __restrict__ cand_inter) {
  __shared__ float row[R_];
  int l = blockIdx.x;                       // 0..LG_-1
  int b = blockIdx.y;
  const float* src = dwe + ((size_t)b * LG_ + l) * R_;
  for (int k = threadIdx.x; k < R_; k += blockDim.x) row[k] = src[k];
  __syncthreads();
  float x2 = dwe_n2[b * LG_ + l];
  int wave = threadIdx.x >> 5, lane = threadIdx.x & 31;
  for (int j = wave; j < 22; j += 8) {      // 22 targets over 8 waves
    const float* e; float y2;
    if (j == 0)      { e = doce + (size_t)b * R_;               y2 = doc_n2[b]; }
    else if (j == 1) { e = sume + (size_t)b * R_;               y2 = sum_n2[b]; }
    else             { int c = j - 2;
                       e = cande + ((size_t)b * C_ + c) * R_;   y2 = cand_n2[b * C_ + c]; }
    float dot = 0.0f;
    for (int k = lane; k < R_; k += 32) dot = fmaf(row[k], e[k], dot);
    for (int off = 16; off > 0; off >>= 1) dot += __shfl_down(dot, off, 32);
    if (lane == 0) {
      float inv = 1.0f / pdist_from(x2, y2, dot);
      if (j == 0)      doc_inter[b * LG_ + l] = inv;
      else if (j == 1) sum_inter[b * LG_ + l] = inv;
      else             cand_inter[((size_t)b * C_ + (j - 2)) * LG_ + l] = inv;
    }
  }
}

// ---------------- final scores ----------------
// blockIdx.x: 0 -> summary_score[b]; 1..20 -> score[b][j-1]
__global__ __launch_bounds__(256)
void final_kernel(const float* __restrict__ doce, const float* __restrict__ doc_n2,
                  const float* __restrict__ sume, const float* __restrict__ sum_n2,
                  const float* __restrict__ cande, const float* __restrict__ cand_n2,
                  const float* __restrict__ doc_inter, const float* __restrict__ sum_inter,
                  const float* __restrict__ cand_inter, float* __restrict__ out) {
  __shared__ float s0[256], s1[256], s2[256], s3[256];
  int j = blockIdx.x;
  int b = blockIdx.y;
  int t = threadIdx.x;
  const float* x; float x2;
  if (j == 0) { x = sume + (size_t)b * R_;                 x2 = sum_n2[b]; }
  else        { x = cande + ((size_t)b * C_ + j - 1) * R_; x2 = cand_n2[b * C_ + j - 1]; }
  const float* y = doce + (size_t)b * R_;
  float y2 = doc_n2[b];

  float xy = 0.0f;
  for (int k = t; k < R_; k += 256) xy = fmaf(x[k], y[k], xy);

  const float* a  = doc_inter + (size_t)b * LG_;
  const float* bv = (j == 0) ? (sum_inter + (size_t)b * LG_)
                             : (cand_inter + ((size_t)b * C_ + j - 1) * LG_);
  float aa = 0.0f, bb = 0.0f, ab = 0.0f;
  for (int k = t; k < LG_; k += 256) {
    float av = a[k], bw = bv[k];
    aa = fmaf(av, av, aa); bb = fmaf(bw, bw, bb); ab = fmaf(av, bw, ab);
  }
  s0[t] = xy; s1[t] = aa; s2[t] = bb; s3[t] = ab;
  __syncthreads();
  for (int off = 128; off > 0; off >>= 1) {
    if (t < off) { s0[t] += s0[t+off]; s1[t] += s1[t+off]; s2[t] += s2[t+off]; s3[t] += s3[t+off]; }
    __syncthreads();
  }
  if (t == 0) {
    float pd  = pdist_from(x2, y2, s0[0]);
    float cos = s3[0] / (fmaxf(sqrtf(s1[0]), 1e-8f) * fmaxf(sqrtf(s2[0]), 1e-8f));
    float res = -pd * pd + cos;
    if (j == 0) out[C_ * B_ + b] = res;          // summary_score after score block
    else        out[b * C_ + (j - 1)] = res;     // score[b][c]
  }
}

// ---------------- workspace layout (floats) ----------------
#define OFF_SMAX   0                                   // B*H            = 3072
#define OFF_CMAX   (OFF_SMAX  + B_*H_)                 // B*C*H          = 61440
#define OFF_DOCE   (OFF_CMAX  + B_*C_*H_)              // B*R            = 2048
#define OFF_SUME   (OFF_DOCE  + B_*R_)                 // B*R            = 2048
#define OFF_CANDE  (OFF_SUME  + B_*R_)                 // B*C*R          = 40960
#define OFF_DOCN   (OFF_CANDE + B_*C_*R_)              // B              = 4
#define OFF_SUMN   (OFF_DOCN  + B_)                    // B              = 4
#define OFF_CANDN  (OFF_SUMN  + B_)                    // B*C            = 80
#define OFF_DWEN   (OFF_CANDN + B_*C_)                 // B*LG           = 4076
#define OFF_GRAM   (OFF_DWEN  + B_*LG_)                // B*LG*R         = 2086912
#define OFF_DWE    (OFF_GRAM  + B_*LG_*R_)             // B*LG*R         = 2086912
#define OFF_DINT   (OFF_DWE   + B_*LG_*R_)             // B*LG           = 4076
#define OFF_SINT   (OFF_DINT  + B_*LG_)                // B*LG           = 4076
#define OFF_CINT   (OFF_SINT  + B_*LG_)                // B*C*LG         = 81520

extern "C" void kernel_launch(void* const* d_in, const int* in_sizes, int n_in,
                              void* d_out, int out_size, void* d_ws, size_t ws_size,
                              hipStream_t stream) {
  (void)in_sizes; (void)n_in; (void)out_size; (void)ws_size;
  const float* text_out      = (const float*)d_in[0];   // (B,L,H)
  const float* summary_out   = (const float*)d_in[1];   // (B,Ls,H)
  const float* candidate_out = (const float*)d_in[2];   // (B,C,Lc,H)
  const float* W_d           = (const float*)d_in[3];   // (H,R)
  const float* W_s           = (const float*)d_in[4];   // (H,R)
  const float* W_c           = (const float*)d_in[5];   // (H,R)
  const float* W_p           = (const float*)d_in[6];   // (R,R)
  const float* conv_w1       = (const float*)d_in[7];   // (1,H,R)
  const float* conv_b1       = (const float*)d_in[8];   // (R,)
  const float* conv_w2       = (const float*)d_in[9];   // (2,H,R) == (1536,R)
  const float* conv_b2       = (const float*)d_in[10];  // (R,)
  float* out = (float*)d_out;
  float* w   = (float*)d_ws;

  float* smax   = w + OFF_SMAX;
  float* cmax   = w + OFF_CMAX;
  float* doce   = w + OFF_DOCE;
  float* sume   = w + OFF_SUME;
  float* cande  = w + OFF_CANDE;
  float* docn   = w + OFF_DOCN;
  float* sumn   = w + OFF_SUMN;
  float* candn  = w + OFF_CANDN;
  float* dwen   = w + OFF_DWEN;
  float* gram   = w + OFF_GRAM;
  float* dwe    = w + OFF_DWE;
  float* dint   = w + OFF_DINT;
  float* sint   = w + OFF_SINT;
  float* cint   = w + OFF_CINT;

  // 1) length-dim max reductions
  {
    int tot = B_ * H_;
    max_reduce_kernel<<<(tot + 255) / 256, 256, 0, stream>>>(summary_out, smax, B_, LS_, H_);
  }
  {
    int tot = B_ * C_ * H_;
    max_reduce_kernel<<<(tot + 255) / 256, 256, 0, stream>>>(candidate_out, cmax, B_ * C_, LC_, H_);
  }

  // 2) projection GEMMs (linear parts), all N = 512 -> grid.y = 8 (4 waves x 16)
  // doc: rows are batches (row stride L*H), M = 4
  wmma_gemm_f32<<<dim3(1, 8, 1), 128, 0, stream>>>(
      text_out, 0, L_ * H_, W_d, R_, nullptr, doce, 0, R_, B_, H_, 0);
  // summary: smax (4 x 768)
  wmma_gemm_f32<<<dim3(1, 8, 1), 128, 0, stream>>>(
      smax, 0, H_, W_s, R_, nullptr, sume, 0, R_, B_, H_, 0);
  // candidates: cmax (80 x 768)
  wmma_gemm_f32<<<dim3(5, 8, 1), 128, 0, stream>>>(
      cmax, 0, H_, W_c, R_, nullptr, cande, 0, R_, B_ * C_, H_, 0);

  // 3) gram: y1 then y2 (two-tap conv as single K=1536 GEMM, rows overlap)
  wmma_gemm_f32<<<dim3(32, 8, B_), 128, 0, stream>>>(
      text_out + H_, (long long)L_ * H_, H_, conv_w1, R_, conv_b1,
      gram, (long long)LG_ * R_, R_, 510, H_, 1);
  wmma_gemm_f32<<<dim3(32, 8, B_), 128, 0, stream>>>(
      text_out + H_, (long long)L_ * H_, H_, conv_w2, R_, conv_b2,
      gram + (size_t)510 * R_, (long long)LG_ * R_, R_, 509, 2 * H_, 1);

  // 4) doc_word linear: gram @ W_p
  wmma_gemm_f32<<<dim3(64, 8, B_), 128, 0, stream>>>(
      gram, (long long)LG_ * R_, R_, W_p, R_, nullptr,
      dwe, (long long)LG_ * R_, R_, LG_, R_, 0);

  // 5) expmap0 (in place) + row norms
  expmap0_kernel<<<B_,          256, 0, stream>>>(doce,  doce,  docn,  R_);
  expmap0_kernel<<<B_,          256, 0, stream>>>(sume,  sume,  sumn,  R_);
  expmap0_kernel<<<B_ * C_,     256, 0, stream>>>(cande, cande, candn, R_);
  expmap0_kernel<<<B_ * LG_,    256, 0, stream>>>(dwe,   dwe,   dwen,  R_);

  // 6) per-token inverse distances vs doc / summary / 20 candidates
  distances_kernel<<<dim3(LG_, B_), 256, 0, stream>>>(
      dwe, dwen, doce, docn, sume, sumn, cande, candn, dint, sint, cint);

  // 7) final scores: out[0..79] = score (4,20); out[80..83] = summary_score
  final_kernel<<<dim3(C_ + 1, B_), 256, 0, stream>>>(
      doce, docn, sume, sumn, cande, candn, dint, sint, cint, out);
}